// BallPointQuery_63256278335594
// MI455X (gfx1250) — compile-verified
//
#include <hip/hip_runtime.h>
#include <hip/hip_bf16.h>
#include <stdint.h>

// Ball query (PointNet++): d2 = |c|^2 + |p|^2 - 2 c.p computed as a single
// V_WMMA_F32_16X16X4_F32 per 16x16 (centroid x point) tile:
//   A row m  = [-2cx, -2cy, -2cz, c2]      (16x4, K=4)
//   B col n  = [ px,   py,   pz,  1 ]^T    (4x16)
//   C        = p2[n] broadcast per column  (accumulator preload)
// One wave owns 16 centroids; lanes 0-15 each own one centroid's compaction.

typedef __attribute__((ext_vector_type(2))) float v2f;
typedef __attribute__((ext_vector_type(8))) float v8f;

#define BQ_RADIUS2 0.04f
#define BQ_MAXS    64

__global__ __launch_bounds__(256)
void BallPointQuery_wmma_kernel(const float* __restrict__ ptc,   // [B,3,N]
                                const float* __restrict__ cent,  // [B,3,M]
                                int* __restrict__ out,           // [B,M,64]
                                int B, int N, int M)
{
    const int lane = threadIdx.x & 31;
    const int wave = threadIdx.x >> 5;
    const int tilesPerB = M >> 4;
    const int tile = blockIdx.x * (blockDim.x >> 5) + wave;
    if (tile >= B * tilesPerB) return;          // never taken with exact grid
    const int b  = tile / tilesPerB;
    const int tm = tile % tilesPerB;

    const float* px = ptc + (size_t)b * 3 * N;
    const float* py = px + N;
    const float* pz = py + N;
    const float* cx = cent + (size_t)b * 3 * M;
    const float* cy = cx + M;
    const float* cz = cy + M;

    const int  mloc = lane & 15;
    const int  mg   = tm * 16 + mloc;
    const bool lo   = lane < 16;

    const float cxm = cx[mg], cym = cy[mg], czm = cz[mg];
    const float c2  = cxm * cxm + cym * cym + czm * czm;

    // A matrix (16x4 f32): VGPR0 = K0 (lanes 0-15) / K2 (lanes 16-31),
    //                      VGPR1 = K1 (lanes 0-15) / K3 (lanes 16-31)
    v2f amat;
    amat.x = lo ? (-2.0f * cxm) : (-2.0f * czm);
    amat.y = lo ? (-2.0f * cym) : c2;

    int  cnt   = lo ? 0 : BQ_MAXS;   // upper half-wave pretends "done"
    int  first = 0;
    int* orow  = out + (((size_t)b * M + mg) << 6);

    for (int base = 0; base < N; base += 16) {
        const int   n  = base + mloc;
        const float x  = px[n];
        const float y  = py[n];
        const float z  = pz[n];
        const float p2 = x * x + y * y + z * z;

        // B matrix (4x16 f32): column n lives in lane n / lane n+16,
        // VGPR0 = K0 / K2, VGPR1 = K1 / K3
        v2f bmat;
        bmat.x = lo ? x : z;
        bmat.y = lo ? y : 1.0f;

        // C preload: p2[n] in every accumulator VGPR (n = lane & 15)
        v8f c = {p2, p2, p2, p2, p2, p2, p2, p2};

        v8f d = __builtin_amdgcn_wmma_f32_16x16x4_f32(
            /*neg_a=*/false, amat, /*neg_b=*/false, bmat,
            /*c_mod=*/(short)0, c, /*reuse_a=*/false, /*reuse_b=*/false);

        // One ballot per accumulator VGPR: bits[15:0] = row v, bits[31:16] = row v+8
        unsigned rowmask = 0u;
#pragma unroll
        for (int v = 0; v < 8; ++v) {
            unsigned bal = __builtin_amdgcn_ballot_w32(d[v] <= BQ_RADIUS2);
            rowmask = ((lane & 7) == v) ? bal : rowmask;
        }

        if (lo) {
            unsigned m16 = (lane < 8) ? (rowmask & 0xffffu) : (rowmask >> 16);
            while (m16 && cnt < BQ_MAXS) {
                const int idx = base + __builtin_ctz(m16);
                m16 &= m16 - 1u;
                if (cnt == 0) first = idx;
                orow[cnt] = idx;
                ++cnt;
            }
        }

        // wave-uniform early exit once all 16 centroids are full
        if (__builtin_amdgcn_ballot_w32(cnt >= BQ_MAXS) == 0xffffffffu) break;
    }

    if (lo) {
        for (int j = cnt; j < BQ_MAXS; ++j) orow[j] = first;
    }
}

extern "C" void kernel_launch(void* const* d_in, const int* in_sizes, int n_in,
                              void* d_out, int out_size, void* d_ws, size_t ws_size,
                              hipStream_t stream) {
    const float* ptc  = (const float*)d_in[0];   // [B,3,N] f32
    const float* cent = (const float*)d_in[1];   // [B,3,M] f32
    int*         out  = (int*)d_out;             // [B,M,64] i32

    const int B = 8;
    const int N = in_sizes[0] / (3 * B);         // 16384
    const int M = in_sizes[1] / (3 * B);         // 1024

    const int tiles        = B * (M / 16);       // 512 waves
    const int wavesPerBlk  = 256 / 32;           // 8
    dim3 block(256);
    dim3 grid((tiles + wavesPerBlk - 1) / wavesPerBlk);
    BallPointQuery_wmma_kernel<<<grid, block, 0, stream>>>(ptc, cent, out, B, N, M);
}